// AttentionAggregator_584115552373
// MI455X (gfx1250) — compile-verified
//
#include <hip/hip_runtime.h>
#include <hip/hip_bf16.h>

#define N_NODES 4096
#define D_MODEL 256
#define NHEADS 8
#define DHEAD 32
#define MASK_WORDS_PER_ROW 128   // 4096 bits / 32

typedef __attribute__((ext_vector_type(2))) float v2f;
typedef __attribute__((ext_vector_type(8))) float v8f;

__device__ __forceinline__ v8f wmma4(v2f a, v2f b, v8f c) {
    // V_WMMA_F32_16X16X4_F32 : D = A(16x4) * B(4x16) + C(16x16), full fp32
    return __builtin_amdgcn_wmma_f32_16x16x4_f32(
        /*neg_a=*/false, a, /*neg_b=*/false, b,
        /*c_mod=*/(short)0, c, /*reuse_a=*/false, /*reuse_b=*/false);
}

// ---------------------------------------------------------------- mask build
__global__ void zero_mask_kernel(unsigned int* __restrict__ m, int n) {
    int i = blockIdx.x * blockDim.x + threadIdx.x;
    if (i < n) m[i] = 0u;
}

__global__ void build_mask_kernel(const int* __restrict__ e, int E,
                                  unsigned int* __restrict__ m) {
    int i = blockIdx.x * blockDim.x + threadIdx.x;
    if (i < E) {
        int r = e[i];        // edge_index[0][i] : query row
        int c = e[i + E];    // edge_index[1][i] : key column
        atomicOr(&m[r * MASK_WORDS_PER_ROW + (c >> 5)], 1u << (c & 31));
    }
}

// ------------------------------------------------------- QKV projection GEMM
// out[n][j] = sum_d x[n][d] * W[j][d] + b[j]       (out = x @ W^T + b)
// One wave computes a 16-row x 64-col tile (4 WMMA accumulators).
__global__ void __launch_bounds__(128) proj_qkv_kernel(
    const float* __restrict__ x,
    const float* __restrict__ Wq, const float* __restrict__ bq,
    const float* __restrict__ Wk, const float* __restrict__ bk,
    const float* __restrict__ Wv, const float* __restrict__ bv,
    float* __restrict__ Q, float* __restrict__ K, float* __restrict__ V)
{
    const float* W; const float* b; float* out;
    if (blockIdx.z == 0)      { W = Wq; b = bq; out = Q; }
    else if (blockIdx.z == 1) { W = Wk; b = bk; out = K; }
    else                      { W = Wv; b = bv; out = V; }

    int lane = threadIdx.x & 31;
    int hi   = lane >> 4;            // 0: lanes 0-15, 1: lanes 16-31
    int l    = lane & 15;
    int cg   = (threadIdx.x >> 5) & 3;   // col group 0..3 within block
    int rbase = blockIdx.x * 16;         // 256 row blocks
    int cbase = cg * 64;

    v8f acc[4] = {};
    const float* arow = x + (size_t)(rbase + l) * D_MODEL + 2 * hi;

    for (int k = 0; k < D_MODEL; k += 4) {
        v2f a = *(const v2f*)(arow + k);   // lanes<16: K=k,k+1 ; lanes>=16: K=k+2,k+3
        #pragma unroll
        for (int t = 0; t < 4; t++) {
            const float* brow = W + (size_t)(cbase + t * 16 + l) * D_MODEL + 2 * hi;
            v2f bb = *(const v2f*)(brow + k);
            acc[t] = wmma4(a, bb, acc[t]);
        }
    }

    #pragma unroll
    for (int t = 0; t < 4; t++) {
        int col = cbase + t * 16 + l;
        float bias = b[col];
        #pragma unroll
        for (int i = 0; i < 8; i++) {
            int row = rbase + i + 8 * hi;     // C layout: vgpr i -> M=i (lo) / i+8 (hi)
            out[(size_t)row * D_MODEL + col] = acc[t][i] + bias;
        }
    }
}

// ----------------------------------------------- fused masked flash attention
// One wave handles (head h, 16-query block). Iterates all 256 key blocks.
__global__ void __launch_bounds__(128) attn_kernel(
    const float* __restrict__ Q, const float* __restrict__ K,
    const float* __restrict__ V, const unsigned int* __restrict__ mask,
    float* __restrict__ Y)
{
    __shared__ float plds[4][16 * 18];   // P staging, padded stride 18

    int wid  = threadIdx.x >> 5;
    int lane = threadIdx.x & 31;
    int hi   = lane >> 4;
    int l    = lane & 15;

    int flat  = blockIdx.x * 4 + wid;    // 0..2047
    int h     = flat >> 8;               // head 0..7
    int qbase = (flat & 255) * 16;       // query block base
    int hoff  = h * DHEAD;

    const float scale = 0.1767766952966369f;   // 1/sqrt(32)

    // Q fragment in A-layout, pre-scaled: 8 K-chunks of 4
    v2f qa[8];
    const float* qrow = Q + (size_t)(qbase + l) * D_MODEL + hoff + 2 * hi;
    #pragma unroll
    for (int c = 0; c < 8; c++) {
        v2f t = *(const v2f*)(qrow + 4 * c);
        qa[c] = t * scale;
    }

    float mrun[8], lrun[8];
    v8f acc0 = {}, acc1 = {};
    #pragma unroll
    for (int i = 0; i < 8; i++) { mrun[i] = -__builtin_inff(); lrun[i] = 0.0f; }

    float* P = &plds[wid][0];

    for (int kb = 0; kb < N_NODES / 16; kb++) {
        int kbase = kb * 16;

        // ---- S = (Q*scale) . K^T  (16x16), B[d][m] = K[kbase+m][hoff+d]
        v8f s = {};
        const float* krow = K + (size_t)(kbase + l) * D_MODEL + hoff + 2 * hi;
        #pragma unroll
        for (int c = 0; c < 8; c++) {
            v2f bk_ = *(const v2f*)(krow + 4 * c);
            s = wmma4(qa[c], bk_, s);
        }

        // ---- mask + online softmax (C layout: vgpr i -> row i+8*hi, col l)
        unsigned int wv = mask[(size_t)(qbase + l) * MASK_WORDS_PER_ROW + (kbase >> 5)];
        int bitbase = kbase & 16;
        #pragma unroll
        for (int i = 0; i < 8; i++) {
            unsigned int wrow = __shfl(wv, i + 8 * hi, 16);   // word of row qbase+i+8*hi
            bool ok = (wrow >> (bitbase + l)) & 1u;
            float sv = ok ? s[i] : -__builtin_inff();
            float mx = sv;
            mx = fmaxf(mx, __shfl_xor(mx, 1, 16));
            mx = fmaxf(mx, __shfl_xor(mx, 2, 16));
            mx = fmaxf(mx, __shfl_xor(mx, 4, 16));
            mx = fmaxf(mx, __shfl_xor(mx, 8, 16));
            float mnew = fmaxf(mrun[i], mx);
            float al = (mnew == -__builtin_inff()) ? 0.0f : __expf(mrun[i] - mnew);
            float pv = ok ? __expf(sv - mnew) : 0.0f;
            float ls = pv;
            ls += __shfl_xor(ls, 1, 16);
            ls += __shfl_xor(ls, 2, 16);
            ls += __shfl_xor(ls, 4, 16);
            ls += __shfl_xor(ls, 8, 16);
            lrun[i] = lrun[i] * al + ls;
            mrun[i] = mnew;
            acc0[i] *= al;
            acc1[i] *= al;
            P[(i + 8 * hi) * 18 + l] = pv;   // C layout -> LDS scratch
        }

        // ---- O += P . V_block   (A = P re-read in A layout from LDS)
        const float* vrow = V + (size_t)kbase * D_MODEL + hoff;
        #pragma unroll
        for (int c = 0; c < 4; c++) {
            int k0 = 4 * c + 2 * hi;
            v2f pa;
            pa.x = P[l * 18 + k0 + 0];
            pa.y = P[l * 18 + k0 + 1];
            v2f b0, b1;
            b0.x = vrow[(size_t)(k0 + 0) * D_MODEL + l];
            b0.y = vrow[(size_t)(k0 + 1) * D_MODEL + l];
            b1.x = vrow[(size_t)(k0 + 0) * D_MODEL + 16 + l];
            b1.y = vrow[(size_t)(k0 + 1) * D_MODEL + 16 + l];
            acc0 = wmma4(pa, b0, acc0);
            acc1 = wmma4(pa, b1, acc1);
        }
    }

    // ---- finalize: y[row][hoff + col] = acc / l   (self-loop => lrun > 0)
    #pragma unroll
    for (int i = 0; i < 8; i++) {
        float inv = 1.0f / lrun[i];
        int row = qbase + i + 8 * hi;
        Y[(size_t)row * D_MODEL + hoff + l]      = acc0[i] * inv;
        Y[(size_t)row * D_MODEL + hoff + 16 + l] = acc1[i] * inv;
    }
}

// -------------------------------------------- final projection over [x | y]
// out[n][j] = sum_{d<512} cat[n][d] * Wp[j][d] + bp[j]
__global__ void __launch_bounds__(128) proj_out_kernel(
    const float* __restrict__ x, const float* __restrict__ Y,
    const float* __restrict__ Wp, const float* __restrict__ bp,
    float* __restrict__ out)
{
    int lane = threadIdx.x & 31;
    int hi   = lane >> 4;
    int l    = lane & 15;
    int cg   = (threadIdx.x >> 5) & 3;
    int rbase = blockIdx.x * 16;
    int cbase = cg * 64;

    v8f acc[4] = {};
    const float* ax = x + (size_t)(rbase + l) * D_MODEL + 2 * hi;
    const float* ay = Y + (size_t)(rbase + l) * D_MODEL + 2 * hi;

    for (int k = 0; k < 2 * D_MODEL; k += 4) {
        v2f a = (k < D_MODEL) ? *(const v2f*)(ax + k)
                              : *(const v2f*)(ay + (k - D_MODEL));
        #pragma unroll
        for (int t = 0; t < 4; t++) {
            const float* brow = Wp + (size_t)(cbase + t * 16 + l) * (2 * D_MODEL) + 2 * hi;
            v2f bb = *(const v2f*)(brow + k);
            acc[t] = wmma4(a, bb, acc[t]);
        }
    }

    #pragma unroll
    for (int t = 0; t < 4; t++) {
        int col = cbase + t * 16 + l;
        float bias = bp[col];
        #pragma unroll
        for (int i = 0; i < 8; i++) {
            int row = rbase + i + 8 * hi;
            out[(size_t)row * D_MODEL + col] = acc[t][i] + bias;
        }
    }
}

// ---------------------------------------------------------------------------
extern "C" void kernel_launch(void* const* d_in, const int* in_sizes, int n_in,
                              void* d_out, int out_size, void* d_ws, size_t ws_size,
                              hipStream_t stream) {
    const float* x   = (const float*)d_in[0];
    const int*   ei  = (const int*)d_in[1];
    const float* Wq  = (const float*)d_in[2];
    const float* bq  = (const float*)d_in[3];
    const float* Wk  = (const float*)d_in[4];
    const float* bk  = (const float*)d_in[5];
    const float* Wv  = (const float*)d_in[6];
    const float* bv  = (const float*)d_in[7];
    const float* Wp  = (const float*)d_in[8];
    const float* bp  = (const float*)d_in[9];
    float* out = (float*)d_out;

    int E = in_sizes[1] / 2;

    // workspace layout: Q | K | V | Y (floats), then mask (uint32)
    float* Q = (float*)d_ws;
    float* K = Q + (size_t)N_NODES * D_MODEL;
    float* V = K + (size_t)N_NODES * D_MODEL;
    float* Y = V + (size_t)N_NODES * D_MODEL;
    unsigned int* mask = (unsigned int*)(Y + (size_t)N_NODES * D_MODEL);
    int mask_words = N_NODES * MASK_WORDS_PER_ROW;

    zero_mask_kernel<<<(mask_words + 255) / 256, 256, 0, stream>>>(mask, mask_words);
    build_mask_kernel<<<(E + 255) / 256, 256, 0, stream>>>(ei, E, mask);

    dim3 gproj(N_NODES / 16, 1, 3);
    proj_qkv_kernel<<<gproj, 128, 0, stream>>>(x, Wq, bq, Wk, bk, Wv, bv, Q, K, V);

    // 8 heads * 256 query blocks = 2048 waves; 4 waves per block
    attn_kernel<<<(NHEADS * (N_NODES / 16)) / 4, 128, 0, stream>>>(Q, K, V, mask, Y);

    proj_out_kernel<<<N_NODES / 16, 128, 0, stream>>>(x, Y, Wp, bp, out);
}